// YOLOV3TargetGenerator_59227599012159
// MI455X (gfx1250) — compile-verified
//
#include <hip/hip_runtime.h>

// ---------------- fixed problem geometry (from reference setup) ----------------
#define NCLS   80
#define BATCH  16
#define MGT    50            // gt boxes per image (last 10 are invalid -1 rows)
#define NPRED  22743         // 361*3 + 1444*3 + 5776*3
#define BN     (BATCH * NPRED)   // 363888 (divisible by 4)

// native clang vector type (required by __builtin_nontemporal_store)
typedef float v4f __attribute__((ext_vector_type(4)));

// d_out regions (float offsets):
//   obj   : [0      , 1*BN)
//   center: [1*BN   , 3*BN)
//   scale : [3*BN   , 5*BN)
//   weight: [5*BN   , 7*BN)
//   cls   : [7*BN   , 87*BN)   default -1
//   cmask : [87*BN  , 167*BN)  default 0

// -------- Kernel A: dynamic objectness = -1 where max IoU(pred, any gt) > 0.5 --------
__global__ __launch_bounds__(256)
void obj_iou_kernel(const float* __restrict__ box_preds,
                    const float* __restrict__ gt_boxes,
                    float* __restrict__ obj_out) {
    __shared__ float4 sgt[MGT];
    const int b = blockIdx.y;
    const int t = threadIdx.x;

    // Async DMA the 50 gt boxes (800 B) straight into LDS: memory -> LDS with no
    // VGPR round-trip, tracked on ASYNCcnt (CDNA5 GLOBAL_LOAD_ASYNC_TO_LDS_B128).
    if (t < MGT) {
        const unsigned lds_addr = (unsigned)(size_t)(&sgt[0]) + (unsigned)t * 16u;
        const unsigned voff     = (unsigned)t * 16u;
        const float*   src      = gt_boxes + (size_t)b * MGT * 4;
        asm volatile("global_load_async_to_lds_b128 %0, %1, %2"
                     :: "v"(lds_addr), "v"(voff), "s"(src)
                     : "memory");
    }
    asm volatile("s_wait_asynccnt 0x0" ::: "memory");
    __syncthreads();

    const int n = blockIdx.x * blockDim.x + t;
    if (n >= NPRED) return;

    const float4* pp = (const float4*)box_preds + ((long long)b * NPRED + n);
    __builtin_prefetch(pp, 0, 0);                 // global_prefetch_b8
    const float4 p = *pp;                         // global_load_b128
    const float a1 = (p.z - p.x) * (p.w - p.y);

    float best = 0.0f;
    #pragma unroll 10
    for (int m = 0; m < MGT; ++m) {
        const float4 g = sgt[m];                  // ds_load_b128
        float iw = fminf(p.z, g.z) - fmaxf(p.x, g.x);
        float ih = fminf(p.w, g.w) - fmaxf(p.y, g.y);
        iw = fmaxf(iw, 0.0f);
        ih = fmaxf(ih, 0.0f);
        const float inter = iw * ih;
        const float a2 = (g.z - g.x) * (g.w - g.y);
        const float uni = fmaxf(a1 + a2 - inter, 1e-6f);
        best = fmaxf(best, inter / uni);
    }
    obj_out[(long long)b * NPRED + n] = (best > 0.5f) ? -1.0f : 0.0f;
}

// -------- Kernel B: stream the constant defaults with non-temporal b128 stores --------
__global__ __launch_bounds__(256)
void fill_defaults_kernel(v4f* __restrict__ out4) {
    // float4-index space covering floats [BN, 167*BN)
    const long long total    = (166LL * BN) / 4;
    const long long clsBeg   = (6LL   * BN) / 4;   // relative: cls region start
    const long long clsEnd   = (86LL  * BN) / 4;   // relative: cls region end
    const long long base     = BN / 4;
    const long long stride   = (long long)gridDim.x * blockDim.x;
    for (long long i = (long long)blockIdx.x * blockDim.x + threadIdx.x;
         i < total; i += stride) {
        const float s = (i >= clsBeg && i < clsEnd) ? -1.0f : 0.0f;
        v4f v;
        v.x = s; v.y = s; v.z = s; v.w = s;
        // write-once streaming data larger than L2 -> non-temporal b128 store
        __builtin_nontemporal_store(v, out4 + base + i);
    }
}

// -------- Kernel C: per-image sparse scatter of matched GT targets --------
__global__ __launch_bounds__(64)
void scatter_kernel(const float* __restrict__ gt_boxes,
                    const int*   __restrict__ gt_labels,
                    const float* __restrict__ anchors,
                    const int*   __restrict__ pad_h_p,
                    const int*   __restrict__ pad_w_p,
                    float* __restrict__ out) {
    __shared__ float sanc[18];
    __shared__ int   sok[MGT];
    __shared__ int   sn[MGT];

    const int b = blockIdx.x;
    const int t = threadIdx.x;

    if (t < 18) sanc[t] = anchors[t];

    float4 g = make_float4(0.f, 0.f, 0.f, 0.f);
    if (t < MGT) {
        g = ((const float4*)gt_boxes)[b * MGT + t];
        sok[t] = (g.x >= 0.0f) && (g.y >= 0.0f) && (g.z >= 0.0f) && (g.w >= 0.0f);
    }
    __syncthreads();

    int   n_out = -1, match = 0, locx = 0, locy = 0;
    float fx = 0.f, fy = 0.f, sw = 0.f, sh = 0.f, wgt = 0.f;

    if (t < MGT) {
        // valid = prefix-AND of per-gt "all coords >= 0" (cumprod in reference)
        int valid = 1;
        for (int j = 0; j <= t; ++j) valid &= sok[j];

        if (valid) {
            const float gtw = g.z - g.x, gth = g.w - g.y;
            const float gtx = 0.5f * (g.x + g.z), gty = 0.5f * (g.y + g.w);
            const float garea = gtw * gth;

            // shape IoU vs 9 anchors (both centered at origin); argmax = first max
            float best = -1.0f;
            #pragma unroll
            for (int a = 0; a < 9; ++a) {
                const float aw = sanc[2 * a], ah = sanc[2 * a + 1];
                const float inter = fminf(aw, gtw) * fminf(ah, gth);
                const float uni   = fmaxf(aw * ah + garea - inter, 1e-6f);
                const float iou   = inter / uni;
                if (iou > best) { best = iou; match = a; }
            }
            const int nl = match / 3;                         // searchsorted([3,6,9], right)
            const int W  = (nl == 0) ? 19 : ((nl == 1) ? 38 : 76);
            const int nbase = (nl == 0) ? 0 : ((nl == 1) ? 1083 : 5415);
            const float padw = (float)pad_w_p[0];
            const float padh = (float)pad_h_p[0];
            fx = gtx / padw * (float)W;
            fy = gty / padh * (float)W;
            locx = (int)fx;                                    // trunc == floor (fx >= 0)
            locy = (int)fy;
            n_out = nbase + (locy * W + locx) * 3 + (match - 3 * nl);
            sw  = logf(fmaxf(gtw, 1.0f) / sanc[2 * match]);
            sh  = logf(fmaxf(gth, 1.0f) / sanc[2 * match + 1]);
            wgt = 2.0f - garea / (padw * padh);
        }
    }
    if (t < MGT) sn[t] = n_out;
    __syncthreads();

    if (t < MGT && n_out >= 0) {
        // last-write-wins: survive iff no later valid gt targets the same output n
        bool win = true;
        for (int j = t + 1; j < MGT; ++j)
            if (sn[j] == n_out) win = false;
        if (win) {
            const long long base = (long long)b * NPRED + n_out;
            out[base] = 1.0f;                                  // objectness
            float* cen = out + 1LL * BN;
            cen[2 * base + 0] = fx - (float)locx;
            cen[2 * base + 1] = fy - (float)locy;
            float* sca = out + 3LL * BN;
            sca[2 * base + 0] = sw;
            sca[2 * base + 1] = sh;
            float* wg = out + 5LL * BN;
            wg[2 * base + 0] = wgt;
            wg[2 * base + 1] = wgt;
            const int lab = gt_labels[b * MGT + t] - 1;        // one_hot(label-1)
            float* cls = out + 7LL * BN + 80LL * base;
            float* cm  = out + 87LL * BN + 80LL * base;
            #pragma unroll 8
            for (int c = 0; c < NCLS; ++c) {
                cls[c] = (c == lab) ? 1.0f : 0.0f;
                cm[c]  = 1.0f;                                 // mask==1 on matched cells
            }
        }
    }
}

extern "C" void kernel_launch(void* const* d_in, const int* in_sizes, int n_in,
                              void* d_out, int out_size, void* d_ws, size_t ws_size,
                              hipStream_t stream) {
    const float* box_preds = (const float*)d_in[0];   // (B,N,4) f32
    const float* gt_boxes  = (const float*)d_in[1];   // (B,M,4) f32
    const float* anchors   = (const float*)d_in[2];   // (9,2)   f32
    const int*   gt_labels = (const int*)  d_in[3];   // (B,M)   i32
    const int*   pad_h     = (const int*)  d_in[4];   // scalar
    const int*   pad_w     = (const int*)  d_in[5];   // scalar
    float* out = (float*)d_out;

    (void)in_sizes; (void)n_in; (void)out_size; (void)d_ws; (void)ws_size;

    // A: dynamic objectness (disjoint region from B)
    dim3 gA((NPRED + 255) / 256, BATCH);
    obj_iou_kernel<<<gA, 256, 0, stream>>>(box_preds, gt_boxes, out);

    // B: constant defaults for centers/scales/weights/cls/cmask
    fill_defaults_kernel<<<2048, 256, 0, stream>>>((v4f*)out);

    // C: sparse overwrite at matched positions (must follow A and B; same stream)
    scatter_kernel<<<BATCH, 64, 0, stream>>>(gt_boxes, gt_labels, anchors,
                                             pad_h, pad_w, out);
}